// SimpleTestGCN_23321672417513
// MI455X (gfx1250) — compile-verified
//
#include <hip/hip_runtime.h>

// ---------------------------------------------------------------------------
// GCN (in_dim=1) on MI455X. Rank-1 reduction:
//   h = x @ W_gcn  ==>  h[i,k] = x[i]*Wg[k]
//   agg[c,k] = S[c]*Wg[k] + bg[k],  S[c] = sum_e norm_e * x[row_e]  (+self loop)
//   out[c]   = x[c] + bp + sum_k relu(S[c]*Wg[k]+bg[k]) * Wp[k]
// Passes:
//   K1: deg = 1                         (self loops)
//   K2: deg[col_e] += 1                 (streaming b128 + L2 atomics)
//   K3: dinv = rsqrt(deg); S = dinv^2*x (self-loop term)
//   K4: S[col] += dinv[row]*dinv[col]*x[row]
//   K5: WMMA epilogue (V_WMMA_F32_16X16X4_F32 x8 per 16 nodes)
// ---------------------------------------------------------------------------

typedef __attribute__((ext_vector_type(2))) float v2f;
typedef __attribute__((ext_vector_type(8))) float v8f;

#define HID 32

__global__ void gcn_init_deg(float* __restrict__ deg, int n) {
  int i = blockIdx.x * blockDim.x + threadIdx.x;
  if (i < n) deg[i] = 1.0f;
}

__global__ void gcn_count_deg(const int* __restrict__ col,
                              float* __restrict__ deg, int e) {
  int i = blockIdx.x * blockDim.x + threadIdx.x;
  int base = i * 4;
  if (base + 3 < e) {
    // prefetch the stream ~4KB ahead (global_prefetch_b8); speculative, safe.
    __builtin_prefetch(col + base + 1024, 0, 1);
    int4 c = ((const int4*)col)[i];          // coalesced b128 stream
    atomicAdd(&deg[c.x], 1.0f);
    atomicAdd(&deg[c.y], 1.0f);
    atomicAdd(&deg[c.z], 1.0f);
    atomicAdd(&deg[c.w], 1.0f);
  } else {
    for (int j = base; j < e; ++j) atomicAdd(&deg[col[j]], 1.0f);
  }
}

__global__ void gcn_dinv_selfloop(const float* __restrict__ x,
                                  float* __restrict__ deg_dinv,
                                  float* __restrict__ s, int n) {
  int i = blockIdx.x * blockDim.x + threadIdx.x;
  if (i < n) {
    float d = rsqrtf(deg_dinv[i]);   // deg >= 1 always (self loop)
    deg_dinv[i] = d;                 // becomes dinv[]
    s[i] = d * d * x[i];             // self-loop contribution
  }
}

__global__ void gcn_scatter(const int* __restrict__ rows,
                            const int* __restrict__ cols,
                            const float* __restrict__ x,
                            const float* __restrict__ dinv,
                            float* __restrict__ s, int e) {
  int i = blockIdx.x * blockDim.x + threadIdx.x;
  int base = i * 4;
  if (base + 3 < e) {
    __builtin_prefetch(rows + base + 1024, 0, 1);
    __builtin_prefetch(cols + base + 1024, 0, 1);
    int4 r = ((const int4*)rows)[i];         // coalesced b128 streams
    int4 c = ((const int4*)cols)[i];
    atomicAdd(&s[c.x], dinv[r.x] * dinv[c.x] * x[r.x]);
    atomicAdd(&s[c.y], dinv[r.y] * dinv[c.y] * x[r.y]);
    atomicAdd(&s[c.z], dinv[r.z] * dinv[c.z] * x[r.z]);
    atomicAdd(&s[c.w], dinv[r.w] * dinv[c.w] * x[r.w]);
  } else {
    for (int j = base; j < e; ++j) {
      int rr = rows[j], cc = cols[j];
      atomicAdd(&s[cc], dinv[rr] * dinv[cc] * x[rr]);
    }
  }
}

// Epilogue: per wave32, 16 nodes. delta = relu(S*Wg+bg) @ Wp via eight
// chained V_WMMA_F32_16X16X4_F32. B replicates Wp across all 16 columns, so
// every column of D equals the delta vector; lanes 0 and 16 write results.
__global__ void gcn_epilogue_wmma(const float* __restrict__ x,
                                  const float* __restrict__ s,
                                  const float* __restrict__ Wg,
                                  const float* __restrict__ bg,
                                  const float* __restrict__ Wp,
                                  const float* __restrict__ bp,
                                  float* __restrict__ out, int n) {
  __shared__ float sWg[HID], sbg[HID], sWp[HID];
  int tid = threadIdx.x;
  if (tid < HID) {
    sWg[tid] = Wg[tid];
    sbg[tid] = bg[tid];
    sWp[tid] = Wp[tid];
  }
  __syncthreads();

  int wave = tid >> 5;               // wave32
  int lane = tid & 31;
  int base = (blockIdx.x * (blockDim.x >> 5) + wave) * 16;
  if (base >= n) return;             // wave-uniform

  float bpred = bp[0];

  if (base + 16 <= n) {
    // A layout (32-bit 16x4): lanes 0-15 -> M=lane, K={0,1}; lanes 16-31 -> K={2,3}.
    // B layout (4x16): VGPR0 = row K=0 (lanes 0-15) / K=2 (lanes 16-31); VGPR1 = K=1 / K=3.
    int m = lane & 15;
    int khalf = (lane < 16) ? 0 : 2;
    float sv = s[base + m];
    v8f acc = {};
#pragma unroll
    for (int c = 0; c < 8; ++c) {
      int k0 = c * 4 + khalf;
      float t0 = sv * sWg[k0] + sbg[k0];
      float t1 = sv * sWg[k0 + 1] + sbg[k0 + 1];
      v2f a, b;
      a.x = t0 > 0.0f ? t0 : 0.0f;
      a.y = t1 > 0.0f ? t1 : 0.0f;
      b.x = sWp[k0];
      b.y = sWp[k0 + 1];
      acc = __builtin_amdgcn_wmma_f32_16x16x4_f32(
          /*neg_a=*/false, a, /*neg_b=*/false, b,
          /*c_mod=*/(short)0, acc, /*reuse_a=*/false, /*reuse_b=*/false);
    }
    // D layout: VGPR r, lanes 0-15 hold M=r; lanes 16-31 hold M=8+r. All N
    // columns identical, so lane 0 writes nodes base+0..7, lane 16 writes 8..15.
    if ((lane & 15) == 0) {
      int moff = (lane >> 4) * 8;
#pragma unroll
      for (int r = 0; r < 8; ++r) {
        int node = base + moff + r;
        out[node] = x[node] + bpred + acc[r];
      }
    }
  } else {
    // Scalar tail for a partial final group (wave-uniform path; no WMMA here).
    if (lane < 16 && base + lane < n) {
      int node = base + lane;
      float sv = s[node];
      float d = 0.0f;
      for (int k = 0; k < HID; ++k) {
        float t = sv * sWg[k] + sbg[k];
        d += (t > 0.0f ? t : 0.0f) * sWp[k];
      }
      out[node] = x[node] + bpred + d;
    }
  }
}

extern "C" void kernel_launch(void* const* d_in, const int* in_sizes, int n_in,
                              void* d_out, int out_size, void* d_ws, size_t ws_size,
                              hipStream_t stream) {
  const float* x    = (const float*)d_in[0];
  const int*   edge = (const int*)d_in[1];   // [2,E] flattened: rows then cols
  const float* Wg   = (const float*)d_in[2]; // [1,32]
  const float* bg   = (const float*)d_in[3]; // [32]
  const float* Wp   = (const float*)d_in[4]; // [32,1]
  const float* bp   = (const float*)d_in[5]; // [1]
  float* out = (float*)d_out;

  const int n = in_sizes[0];          // N nodes
  const int e = in_sizes[1] / 2;      // E edges
  const int* rows = edge;
  const int* cols = edge + e;

  float* deg_dinv = (float*)d_ws;     // N floats: deg, then overwritten by dinv
  float* s        = deg_dinv + n;     // N floats: scalar aggregate

  const int T = 256;
  dim3 blkN((n + T - 1) / T);
  dim3 blkE(((e + 3) / 4 + T - 1) / T);
  dim3 blkEp((n + 127) / 128);        // 8 waves/block * 16 nodes/wave

  gcn_init_deg<<<blkN, T, 0, stream>>>(deg_dinv, n);
  gcn_count_deg<<<blkE, T, 0, stream>>>(cols, deg_dinv, e);
  gcn_dinv_selfloop<<<blkN, T, 0, stream>>>(x, deg_dinv, s, n);
  gcn_scatter<<<blkE, T, 0, stream>>>(rows, cols, x, deg_dinv, s, e);
  gcn_epilogue_wmma<<<blkEp, T, 0, stream>>>(x, s, Wg, bg, Wp, bp, out, n);
}